// EmbeddingFusionLayer_31653908972244
// MI455X (gfx1250) — compile-verified
//
#include <hip/hip_runtime.h>
#include <cstdint>

// ---------------------------------------------------------------------------
// EmbeddingFusionLayer for MI455X (gfx1250, wave32)
//
// users[n,e]    = sum_v softmax(uw)[v] * view_users[v,n,e]     (4 x 1M x 64 f32)
// products[m,e] = sum_v softmax(pw)[v] * view_products[v,m,e]  (4 x 500K x 64 f32)
//
// Bandwidth-bound: 1.536 GB read + 0.384 GB written -> ~82 us floor at
// 23.3 TB/s HBM. Arithmetic intensity ~0.35 FLOP/byte, so WMMA is
// inapplicable (K=4 rank-1 broadcast contraction; 16 useful outputs per
// wmma vs 32 FMAs per VALU instruction). The CDNA5 path that matters is
// async global->LDS staging (GLOBAL_LOAD_ASYNC_TO_LDS_B128 + ASYNCcnt)
// in a depth-2 software pipeline, with B128 traffic end-to-end and
// non-temporal stores to keep the 192 MB L2 clean for the input stream.
// ---------------------------------------------------------------------------

#define N_VIEW    4
#define N_USER    1000000
#define N_PRODUCT 500000
#define EMB       64

#define TILE        256                                  // threads/block; 1 v4f per thread per tile
#define CH_USER     (N_USER * EMB / 4)                   // 16,000,000 float4 chunks
#define CH_PROD     (N_PRODUCT * EMB / 4)                //  8,000,000 float4 chunks
#define TILES_USER  (CH_USER / TILE)                     // 62,500 (exact)
#define TILES_TOTAL ((CH_USER + CH_PROD) / TILE)         // 93,750 (exact, no tail)
#define GRID        3072

typedef float v4f __attribute__((ext_vector_type(4)));   // native clang vector (B128)

// ---- CDNA5 async counter wait -------------------------------------------
// Inline asm (not the builtin) so the "memory" clobber is a guaranteed
// compiler fence: the ds_load readbacks must not be hoisted above the wait.
#define WAIT_ASYNC(n) asm volatile("s_wait_asynccnt %0" ::"i"(n) : "memory")

// ---- CDNA5 async global -> LDS B128 (GV mode), per-lane LDS addr in VGPR --
__device__ __forceinline__ void async_ld_b128(uint32_t lds_off, const v4f* g) {
  asm volatile("global_load_async_to_lds_b128 %0, %1, off"
               :: "v"(lds_off), "v"(g)
               : "memory");
}

__device__ __forceinline__ void softmax4(const float* __restrict__ w, float o[4]) {
  float a = w[0], b = w[1], c = w[2], d = w[3];
  float m  = fmaxf(fmaxf(a, b), fmaxf(c, d));
  float e0 = expf(a - m), e1 = expf(b - m), e2 = expf(c - m), e3 = expf(d - m);
  float inv = 1.0f / (e0 + e1 + e2 + e3);
  o[0] = e0 * inv; o[1] = e1 * inv; o[2] = e2 * inv; o[3] = e3 * inv;
}

__global__ void __launch_bounds__(TILE)
fuse_views_kernel(const float* __restrict__ vu, const float* __restrict__ vp,
                  const float* __restrict__ uwraw, const float* __restrict__ pwraw,
                  float* __restrict__ out) {
  // double-buffered LDS staging: 2 * 4 views * 256 lanes * 16 B = 32 KB
  __shared__ v4f buf[2][N_VIEW][TILE];

  const int tid = threadIdx.x;
  const v4f* vu4  = (const v4f*)vu;
  const v4f* vp4  = (const v4f*)vp;
  v4f*       out4 = (v4f*)out;

  float uw[4], pw[4];
  softmax4(uwraw, uw);
  softmax4(pwraw, pw);

  // prefetch: issue 4 async B128 loads (one per view) for tile t into buffer bi
  auto prefetch = [&](int t, int bi) {
    long c = (long)t * TILE + tid;            // global float4 chunk id
    const v4f* base;
    long vs;                                  // view stride in float4 units
    if (t < TILES_USER) { base = vu4 + c;                   vs = CH_USER; }
    else                { base = vp4 + (c - (long)CH_USER); vs = CH_PROD; }
#pragma unroll
    for (int v = 0; v < N_VIEW; ++v) {
      uint32_t lo = (uint32_t)(uintptr_t)&buf[bi][v][tid]; // low 32b of flat = LDS byte offset
      async_ld_b128(lo, base + (long)v * vs);
    }
  };

  int t = blockIdx.x;
  if (t >= TILES_TOTAL) return;
  int bi = 0;
  prefetch(t, bi);

  for (; t < TILES_TOTAL; t += GRID) {
    const int  tn      = t + GRID;
    const bool hasNext = (tn < TILES_TOTAL);

    if (hasNext) {
      prefetch(tn, bi ^ 1);   // keep 8 async loads in flight
      WAIT_ASYNC(4);          // FIFO completion: tile t's 4 loads are done
    } else {
      WAIT_ASYNC(0);
    }

    // weights for this tile
    const bool isU = (t < TILES_USER);
    const float w0 = isU ? uw[0] : pw[0];
    const float w1 = isU ? uw[1] : pw[1];
    const float w2 = isU ? uw[2] : pw[2];
    const float w3 = isU ? uw[3] : pw[3];

    // each lane reads back exactly the LDS slots it queued -> no barrier
    // needed; the FMA's implicit s_wait_dscnt precedes the next-iteration
    // prefetch into this buffer, resolving the WAR on buffer reuse.
    v4f a  = buf[bi][0][tid];
    v4f b  = buf[bi][1][tid];
    v4f c4 = buf[bi][2][tid];
    v4f d  = buf[bi][3][tid];

    v4f r;
    r.x = fmaf(w3, d.x, fmaf(w2, c4.x, fmaf(w1, b.x, w0 * a.x)));
    r.y = fmaf(w3, d.y, fmaf(w2, c4.y, fmaf(w1, b.y, w0 * a.y)));
    r.z = fmaf(w3, d.z, fmaf(w2, c4.z, fmaf(w1, b.z, w0 * a.z)));
    r.w = fmaf(w3, d.w, fmaf(w2, c4.w, fmaf(w1, b.w, w0 * a.w)));

    long c = (long)t * TILE + tid;            // == output float4 index (users||products concat)
    __builtin_nontemporal_store(r, &out4[c]); // streaming B128 store, keep L2 clean

    bi ^= 1;
  }
}

extern "C" void kernel_launch(void* const* d_in, const int* in_sizes, int n_in,
                              void* d_out, int out_size, void* d_ws, size_t ws_size,
                              hipStream_t stream) {
  (void)in_sizes; (void)n_in; (void)out_size; (void)d_ws; (void)ws_size;
  const float* vu = (const float*)d_in[0];   // [4, 1_000_000, 64]
  const float* vp = (const float*)d_in[1];   // [4,   500_000, 64]
  const float* uw = (const float*)d_in[2];   // [1, 4]
  const float* pw = (const float*)d_in[3];   // [1, 4]
  float*       out = (float*)d_out;          // users (64M f32) || products (32M f32)

  fuse_views_kernel<<<GRID, TILE, 0, stream>>>(vu, vp, uw, pw, out);
}